// DynamicSimpleVFE_6614249636210
// MI455X (gfx1250) — compile-verified
//
#include <hip/hip_runtime.h>
#include <stdint.h>

namespace {
constexpr int kGX = 352, kGY = 400, kGZ = 1, kB = 4;
constexpr int kNSeg = kB * kGZ * kGY * kGX;   // 563200
constexpr int kC = 4;
}

// CDNA5 async data mover: global memory -> LDS, tracked by ASYNCcnt.
// lds_addr is the per-lane LDS byte address (low 32 bits of the generic
// shared-aperture pointer), gptr is the 64-bit global address.
// TH_LOAD_NT: point stream is single-use; don't let it displace the
// L2-resident accumulator array that the atomics hammer.
__device__ __forceinline__ void async_b128_to_lds_nt(unsigned lds_addr, const void* gptr) {
  asm volatile("global_load_async_to_lds_b128 %0, %1, off th:TH_LOAD_NT"
               :: "v"(lds_addr), "v"(gptr)
               : "memory");
}

__global__ __launch_bounds__(256) void vfe_zero_kernel(float4* __restrict__ p, int n4) {
  int i = blockIdx.x * blockDim.x + threadIdx.x;
  const int stride = gridDim.x * blockDim.x;
  const float4 z = make_float4(0.f, 0.f, 0.f, 0.f);
  for (; i < n4; i += stride) p[i] = z;
}

__global__ __launch_bounds__(256) void vfe_scatter_kernel(
    const float4* __restrict__ feats, const int4* __restrict__ coors,
    float* __restrict__ sums, float* __restrict__ counts, int npts) {
  // Per-thread: two 32B staging buffers (16B feature + 16B coord each).
  // 256 threads * 64B = 16 KB LDS (WGP has 320 KB).
  __shared__ char smem[256 * 64];
  char* slot = smem + threadIdx.x * 64;
  const unsigned lds_slot = (unsigned)(unsigned long long)(uintptr_t)slot;

  const int tid = blockIdx.x * blockDim.x + threadIdx.x;
  const int stride = gridDim.x * blockDim.x;
  const int niter = (npts + stride - 1) / stride;
  if (niter <= 0) return;

  // Prologue: stage iteration 0 into buffer 0 (clamped address, always valid).
  {
    const int idx0 = tid < npts ? tid : npts - 1;
    async_b128_to_lds_nt(lds_slot + 0u, feats + idx0);
    async_b128_to_lds_nt(lds_slot + 16u, coors + idx0);
  }

  for (int k = 0; k < niter; ++k) {
    const unsigned curOff = (k & 1) ? 32u : 0u;
    if (k + 1 < niter) {
      // Prefetch next tile into the other buffer, then wait until only the
      // in-flight pair remains (ASYNCcnt <= 2) => current pair has landed.
      const unsigned nxtOff = curOff ^ 32u;
      long long nidx = (long long)tid + (long long)(k + 1) * stride;
      if (nidx > (long long)npts - 1) nidx = (long long)npts - 1;
      async_b128_to_lds_nt(lds_slot + nxtOff, feats + nidx);
      async_b128_to_lds_nt(lds_slot + nxtOff + 16u, coors + nidx);
      asm volatile("s_wait_asynccnt 0x2" ::: "memory");
    } else {
      asm volatile("s_wait_asynccnt 0x0" ::: "memory");
    }

    const float4 f = *(const float4*)(slot + curOff);
    const int4  c = *(const int4*)(slot + curOff + 16);

    const long long idx = (long long)tid + (long long)k * stride;
    if (idx < npts) {
      // seg = ((b*GZ + z)*GY + y)*GX + x
      const int seg = ((c.x * kGZ + c.y) * kGY + c.z) * kGX + c.w;
      float* dst = sums + (long long)seg * kC;
      atomicAdd(dst + 0, f.x);   // non-returning global_atomic_add_f32 -> L2
      atomicAdd(dst + 1, f.y);
      atomicAdd(dst + 2, f.z);
      atomicAdd(dst + 3, f.w);
      atomicAdd(counts + seg, 1.0f);
    }
  }
}

__global__ __launch_bounds__(256) void vfe_finalize_kernel(
    float4* __restrict__ mean4, const float* __restrict__ counts, int nseg) {
  int i = blockIdx.x * blockDim.x + threadIdx.x;
  const int stride = gridDim.x * blockDim.x;
  for (; i < nseg; i += stride) {
    const float cnt = counts[i];
    const float inv = 1.0f / fmaxf(cnt, 1.0f);
    float4 s = mean4[i];
    s.x *= inv; s.y *= inv; s.z *= inv; s.w *= inv;
    mean4[i] = s;
  }
}

extern "C" void kernel_launch(void* const* d_in, const int* in_sizes, int n_in,
                              void* d_out, int out_size, void* d_ws, size_t ws_size,
                              hipStream_t stream) {
  (void)n_in; (void)d_ws; (void)ws_size;
  const float4* feats = (const float4*)d_in[0];   // (N, 4) f32
  const int4*   coors = (const int4*)d_in[1];     // (N, 4) i32 [b,z,y,x]
  const int npts = in_sizes[0] / kC;

  float* out    = (float*)d_out;
  float* sums   = out;                             // (NSEG, 4) -> becomes mean
  float* counts = out + (size_t)kNSeg * kC;        // (NSEG,)

  // 1) zero-init the whole output (sums + counts); out_size = NSEG*5, /4 exact.
  //    Regular-temporal stores warm the accumulator lines into L2 before the
  //    atomic phase.
  const int n4 = out_size / 4;
  int zb = (n4 + 255) / 256;
  if (zb > 4096) zb = 4096;
  vfe_zero_kernel<<<zb, 256, 0, stream>>>((float4*)out, n4);

  // 2) async-staged atomic scatter: 2048 blocks * 8 wave32 = 16K waves.
  vfe_scatter_kernel<<<2048, 256, 0, stream>>>(feats, coors, sums, counts, npts);

  // 3) in-place mean = sum / max(count, 1).
  const int fb = (kNSeg + 255) / 256;
  vfe_finalize_kernel<<<fb, 256, 0, stream>>>((float4*)sums, counts, kNSeg);
}